// HGNN_59751585022371
// MI455X (gfx1250) — compile-verified
//
#include <hip/hip_runtime.h>
#include <hip/hip_bf16.h>
#include <math.h>

typedef __attribute__((ext_vector_type(2))) float v2f;
typedef __attribute__((ext_vector_type(8))) float v8f;

static constexpr int Nn = 50000;   // nodes per type
static constexpr int Dd = 128;     // feature dim (all layers)
static constexpr int Ee = 600000;  // edges per edge type
static constexpr int TPB = 256;
#define NEG_SLOPE 0.2f

// ---------------------------------------------------------------- utilities
__global__ void zero_f32_kernel(float* __restrict__ p, int n) {
    int i = blockIdx.x * blockDim.x + threadIdx.x;
    if (i < n) p[i] = 0.0f;
}

__global__ void fill_u32_kernel(unsigned* __restrict__ p, unsigned v, int n) {
    int i = blockIdx.x * blockDim.x + threadIdx.x;
    if (i < n) p[i] = v;
}

// -------------------------------------------------- SAGE: scatter mean-agg
// wave-per-edge: 32 lanes x float4 = one 512B feature row, atomic f32 adds.
__global__ void scatter_mean_kernel(const float* __restrict__ xsrc,
                                    const int* __restrict__ ei,  // [2,E]
                                    float* __restrict__ msum,
                                    float* __restrict__ cnt, int E) {
    int gid   = blockIdx.x * blockDim.x + threadIdx.x;
    int lane  = gid & 31;
    int e     = gid >> 5;
    int strid = (gridDim.x * blockDim.x) >> 5;
    for (; e < E; e += strid) {
        int s = ei[e];
        int d = ei[E + e];
        float4 v = reinterpret_cast<const float4*>(xsrc + (size_t)s * Dd)[lane];
        float* mrow = msum + (size_t)d * Dd + lane * 4;
        atomicAdd(mrow + 0, v.x);
        atomicAdd(mrow + 1, v.y);
        atomicAdd(mrow + 2, v.z);
        atomicAdd(mrow + 3, v.w);
        if (lane == 0) atomicAdd(cnt + d, 1.0f);
    }
}

__global__ void mean_div_kernel(float* __restrict__ msum,
                                const float* __restrict__ cnt, int n) {
    int i = blockIdx.x * blockDim.x + threadIdx.x;
    if (i < n) {
        float c = cnt[i >> 7];  // row = i / 128
        msum[i] = msum[i] / fmaxf(c, 1.0f);
    }
}

// ------------------------------------------------------------- WMMA GEMM
// out[r,c] = relu?( alpha*( A1@W1^T + (FUSED? A2@W2^T:0) + bias[c] ) + (addPrev? out:0) )
// A: [nrows,128] row-major, W: [128,128] row-major (out = A @ W^T).
// One block = one 16-row tile staged once in LDS (coalesced b128 sweep);
// wave w computes column tile w (8 waves * 16 = 128 cols).
// FP32 WMMA 16x16x4 fragments (per lane, m=lane&15, kh=lane>>4):
//   A (v2f): a.{x,y} = A[m][k + 2*kh + {0,1}]
//   B (v2f): b.{x,y} = W[col0+n][k + 2*kh + {0,1}],  n = lane&15
//   D (v8f): d[v]    = out[row0 + v + 8*kh][col0 + n]
// Independent accumulator chains (even/odd k-step x matrix chain) break the
// WMMA D->C RAW chain and let the unrolled loop hoist loads past the waits.
template <bool FUSED>
__global__ __launch_bounds__(TPB)
void gemm_kernel(const float* __restrict__ A1, const float* __restrict__ W1,
                 const float* __restrict__ A2, const float* __restrict__ W2,
                 const float* __restrict__ bias,
                 float alpha, int addPrev, int doRelu,
                 float* __restrict__ out, int nrows) {
    __shared__ float As1[16 * Dd];
    __shared__ float As2[FUSED ? 16 * Dd : 4];

    const int t    = threadIdx.x;
    const int row0 = blockIdx.x * 16;
    if (row0 >= nrows) return;  // block-uniform

    // cooperative stage of the 16x128 A tile(s): 512 float4 / 256 threads
    {
        const float4* g1 = reinterpret_cast<const float4*>(A1 + (size_t)row0 * Dd);
        float4*       s1 = reinterpret_cast<float4*>(As1);
        const float4* g2 = FUSED ? reinterpret_cast<const float4*>(A2 + (size_t)row0 * Dd) : nullptr;
        float4*       s2 = reinterpret_cast<float4*>(As2);
#pragma unroll
        for (int i = t; i < 16 * Dd / 4; i += TPB) {
            s1[i] = g1[i];
            if (FUSED) s2[i] = g2[i];
        }
    }
    __syncthreads();

    const int lane = t & 31;
    const int wv   = t >> 5;    // 0..7 -> column tile
    const int m    = lane & 15;
    const int kh   = lane >> 4; // 0/1
    const int col0 = wv * 16;

    const float* as1   = As1 + m * Dd;
    const float* as2   = As2 + m * Dd;
    const float* w1row = W1 + (size_t)(col0 + m) * Dd;
    const float* w2row = FUSED ? W2 + (size_t)(col0 + m) * Dd : nullptr;

    v8f accA = {0.f, 0.f, 0.f, 0.f, 0.f, 0.f, 0.f, 0.f};
    v8f accB = accA, accC = accA, accD = accA;

#pragma unroll
    for (int kk = 0; kk < Dd; kk += 8) {
        const int ka = kk + 2 * kh;      // even -> 8B-aligned b64 ds/global loads
        const int kb = kk + 4 + 2 * kh;
        v2f a0 = *reinterpret_cast<const v2f*>(as1 + ka);
        v2f b0 = *reinterpret_cast<const v2f*>(w1row + ka);
        v2f a1 = *reinterpret_cast<const v2f*>(as1 + kb);
        v2f b1 = *reinterpret_cast<const v2f*>(w1row + kb);
        accA = __builtin_amdgcn_wmma_f32_16x16x4_f32(false, a0, false, b0,
                                                     (short)0, accA, false, false);
        accB = __builtin_amdgcn_wmma_f32_16x16x4_f32(false, a1, false, b1,
                                                     (short)0, accB, false, false);
        if (FUSED) {
            v2f c0 = *reinterpret_cast<const v2f*>(as2 + ka);
            v2f d0 = *reinterpret_cast<const v2f*>(w2row + ka);
            v2f c1 = *reinterpret_cast<const v2f*>(as2 + kb);
            v2f d1 = *reinterpret_cast<const v2f*>(w2row + kb);
            accC = __builtin_amdgcn_wmma_f32_16x16x4_f32(false, c0, false, d0,
                                                         (short)0, accC, false, false);
            accD = __builtin_amdgcn_wmma_f32_16x16x4_f32(false, c1, false, d1,
                                                         (short)0, accD, false, false);
        }
    }

    const float bcol = bias ? bias[col0 + m] : 0.0f;
#pragma unroll
    for (int v = 0; v < 8; ++v) {
        int r = row0 + v + 8 * kh;
        int c = col0 + m;
        size_t idx = (size_t)r * Dd + c;
        float s = accA[v] + accB[v];
        if (FUSED) s += accC[v] + accD[v];
        float val = alpha * (s + bcol);
        if (addPrev) val += out[idx];
        if (doRelu)  val = fmaxf(val, 0.0f);
        out[idx] = val;
    }
}

// ------------------------------------------------------------- GATv2 edges
__device__ __forceinline__ unsigned ord_encode(float f) {
    unsigned u = __float_as_uint(f);
    return (u & 0x80000000u) ? ~u : (u | 0x80000000u);
}
__device__ __forceinline__ float ord_decode(unsigned k) {
    return (k & 0x80000000u) ? __uint_as_float(k & 0x7fffffffu)
                             : __uint_as_float(~k);
}

// pass 1: e = att . leaky_relu(xl[src] + xr[dst]); store e; segment-max into menc
__global__ void gat_edge_score_kernel(const float* __restrict__ xl,
                                      const float* __restrict__ xr,
                                      const float* __restrict__ att,
                                      const int* __restrict__ ei,
                                      float* __restrict__ eedge,
                                      unsigned* __restrict__ menc, int E) {
    int gid   = blockIdx.x * blockDim.x + threadIdx.x;
    int lane  = gid & 31;
    int e     = gid >> 5;
    int strid = (gridDim.x * blockDim.x) >> 5;
    for (; e < E; e += strid) {
        int s = ei[e];
        int d = ei[E + e];
        float4 l = reinterpret_cast<const float4*>(xl + (size_t)s * Dd)[lane];
        float4 r = reinterpret_cast<const float4*>(xr + (size_t)d * Dd)[lane];
        float4 a = reinterpret_cast<const float4*>(att)[lane];
        float z0 = l.x + r.x; z0 = (z0 > 0.f) ? z0 : NEG_SLOPE * z0;
        float z1 = l.y + r.y; z1 = (z1 > 0.f) ? z1 : NEG_SLOPE * z1;
        float z2 = l.z + r.z; z2 = (z2 > 0.f) ? z2 : NEG_SLOPE * z2;
        float z3 = l.w + r.w; z3 = (z3 > 0.f) ? z3 : NEG_SLOPE * z3;
        float sum = z0 * a.x + z1 * a.y + z2 * a.z + z3 * a.w;
        for (int off = 16; off > 0; off >>= 1)
            sum += __shfl_xor(sum, off, 32);
        if (lane == 0) {
            eedge[e] = sum;
            atomicMax(menc + d, ord_encode(sum));
        }
    }
}

// pass 2: ex = exp(e - m[dst]); e <- ex; denom[dst] += ex
__global__ void gat_softmax_kernel(const int* __restrict__ ei,
                                   float* __restrict__ eedge,
                                   const unsigned* __restrict__ menc,
                                   float* __restrict__ denom, int E) {
    int e     = blockIdx.x * blockDim.x + threadIdx.x;
    int strid = gridDim.x * blockDim.x;
    for (; e < E; e += strid) {
        int d = ei[E + e];
        float m  = ord_decode(menc[d]);
        float ex = __expf(eedge[e] - m);
        eedge[e] = ex;
        atomicAdd(denom + d, ex);
    }
}

// pass 3: out[dst] += scale * (ex/denom[dst]) * xl[src]
__global__ void gat_scatter_kernel(const float* __restrict__ xl,
                                   const int* __restrict__ ei,
                                   const float* __restrict__ eedge,
                                   const float* __restrict__ denom,
                                   float scale, float* __restrict__ out, int E) {
    int gid   = blockIdx.x * blockDim.x + threadIdx.x;
    int lane  = gid & 31;
    int e     = gid >> 5;
    int strid = (gridDim.x * blockDim.x) >> 5;
    for (; e < E; e += strid) {
        int s = ei[e];
        int d = ei[E + e];
        float coef = scale * eedge[e] / (denom[d] + 1e-16f);
        float4 l = reinterpret_cast<const float4*>(xl + (size_t)s * Dd)[lane];
        float* orow = out + (size_t)d * Dd + lane * 4;
        atomicAdd(orow + 0, coef * l.x);
        atomicAdd(orow + 1, coef * l.y);
        atomicAdd(orow + 2, coef * l.z);
        atomicAdd(orow + 3, coef * l.w);
    }
}

// out = relu(out + s1*b1 + s2*b2)
__global__ void finalize_kernel(float* __restrict__ out,
                                const float* __restrict__ b1, float s1,
                                const float* __restrict__ b2, float s2, int n) {
    int i = blockIdx.x * blockDim.x + threadIdx.x;
    if (i < n) {
        int c = i & (Dd - 1);
        float v = out[i] + s1 * b1[c];
        if (b2) v += s2 * b2[c];
        out[i] = fmaxf(v, 0.0f);
    }
}

// ---------------------------------------------------------------- host side
struct Ptrs {
    const float *x_a, *x_b;
    // edge-type index: 0=ab, 1=ba, 2=aa
    const float *sWl[3], *sbl[3], *sWr[3];
    const float *gWl[3], *gbl[3], *gWr[3], *gbr[3], *gatt[3], *gb[3];
    const int* ei[3];
};

extern "C" void kernel_launch(void* const* d_in, const int* in_sizes, int n_in,
                              void* d_out, int out_size, void* d_ws, size_t ws_size,
                              hipStream_t stream) {
    Ptrs P;
    const bool sorted = (in_sizes[0] == 2 * Ee);  // jax tree_flatten sorted-key order
    if (!sorted) {
        // insertion order: x_a, x_b, [s_et{Wl,bl,Wr}, g_et{Wl,bl,Wr,br,att,b}] for ab,ba,aa, ei_ab,ei_ba,ei_aa
        P.x_a = (const float*)d_in[0];
        P.x_b = (const float*)d_in[1];
        for (int et = 0; et < 3; ++et) {
            int base = 2 + et * 9;
            P.sWl[et]  = (const float*)d_in[base + 0];
            P.sbl[et]  = (const float*)d_in[base + 1];
            P.sWr[et]  = (const float*)d_in[base + 2];
            P.gWl[et]  = (const float*)d_in[base + 3];
            P.gbl[et]  = (const float*)d_in[base + 4];
            P.gWr[et]  = (const float*)d_in[base + 5];
            P.gbr[et]  = (const float*)d_in[base + 6];
            P.gatt[et] = (const float*)d_in[base + 7];
            P.gb[et]   = (const float*)d_in[base + 8];
            P.ei[et]   = (const int*)d_in[29 + et];
        }
    } else {
        // sorted keys: ei_aa, ei_ab, ei_ba, params{g_aa,g_ab,g_ba,s_aa,s_ab,s_ba}, x_a, x_b
        // g sub-keys sorted: Wl, Wr, att, b, bl, br ; s sub-keys sorted: Wl, Wr, bl
        const int eiIdx[3] = {1, 2, 0};     // et(ab,ba,aa) -> leaf index
        const int gBase[3] = {9, 15, 3};
        const int sBase[3] = {24, 27, 21};
        for (int et = 0; et < 3; ++et) {
            P.ei[et]   = (const int*)d_in[eiIdx[et]];
            P.gWl[et]  = (const float*)d_in[gBase[et] + 0];
            P.gWr[et]  = (const float*)d_in[gBase[et] + 1];
            P.gatt[et] = (const float*)d_in[gBase[et] + 2];
            P.gb[et]   = (const float*)d_in[gBase[et] + 3];
            P.gbl[et]  = (const float*)d_in[gBase[et] + 4];
            P.gbr[et]  = (const float*)d_in[gBase[et] + 5];
            P.sWl[et]  = (const float*)d_in[sBase[et] + 0];
            P.sWr[et]  = (const float*)d_in[sBase[et] + 1];
            P.sbl[et]  = (const float*)d_in[sBase[et] + 2];
        }
        P.x_a = (const float*)d_in[30];
        P.x_b = (const float*)d_in[31];
    }

    // workspace layout (floats)
    float* w = (float*)d_ws;
    float* mean  = w; w += (size_t)Nn * Dd;
    float* xa1   = w; w += (size_t)Nn * Dd;
    float* xb1   = w; w += (size_t)Nn * Dd;
    float* xl    = w; w += (size_t)Nn * Dd;
    float* xr    = w; w += (size_t)Nn * Dd;
    float* cnt   = w; w += Nn;
    unsigned* menc = (unsigned*)w; w += Nn;
    float* denom = w; w += Nn;
    float* eedge = w; w += Ee;

    float* out_a = (float*)d_out;
    float* out_b = out_a + (size_t)Nn * Dd;

    const int gN   = (Nn + TPB - 1) / TPB;          // per-node kernels
    const int gND  = (Nn * Dd + TPB - 1) / TPB;     // per-element kernels
    const int gE   = (Ee + TPB - 1) / TPB;          // per-edge (thread) kernels
    const int gEW  = (Ee * 32 + TPB - 1) / TPB;     // per-edge (wave) kernels
    const int gGemm = Nn / 16;                      // 50000 / 16 = 3125 tiles

    auto run_sage = [&](const float* xs, const float* xd, int et, float* outp,
                        float alpha, int addPrev, int doRelu) {
        zero_f32_kernel<<<gND, TPB, 0, stream>>>(mean, Nn * Dd);
        zero_f32_kernel<<<gN, TPB, 0, stream>>>(cnt, Nn);
        scatter_mean_kernel<<<gEW, TPB, 0, stream>>>(xs, P.ei[et], mean, cnt, Ee);
        mean_div_kernel<<<gND, TPB, 0, stream>>>(mean, cnt, Nn * Dd);
        gemm_kernel<true><<<gGemm, TPB, 0, stream>>>(mean, P.sWl[et], xd, P.sWr[et],
                                                     P.sbl[et], alpha, addPrev, doRelu,
                                                     outp, Nn);
    };

    auto run_gat = [&](const float* xs, const float* xd, int et, float* outp, float scale) {
        gemm_kernel<false><<<gGemm, TPB, 0, stream>>>(xs, P.gWl[et], nullptr, nullptr,
                                                      P.gbl[et], 1.0f, 0, 0, xl, Nn);
        gemm_kernel<false><<<gGemm, TPB, 0, stream>>>(xd, P.gWr[et], nullptr, nullptr,
                                                      P.gbr[et], 1.0f, 0, 0, xr, Nn);
        fill_u32_kernel<<<gN, TPB, 0, stream>>>(menc, 0x007FFFFFu, Nn);  // encode(-inf)
        zero_f32_kernel<<<gN, TPB, 0, stream>>>(denom, Nn);
        gat_edge_score_kernel<<<gEW, TPB, 0, stream>>>(xl, xr, P.gatt[et], P.ei[et],
                                                       eedge, menc, Ee);
        gat_softmax_kernel<<<gE, TPB, 0, stream>>>(P.ei[et], eedge, menc, denom, Ee);
        gat_scatter_kernel<<<gEW, TPB, 0, stream>>>(xl, P.ei[et], eedge, denom,
                                                    scale, outp, Ee);
    };

    // ---------------- layer 0: SAGEConv + hetero-mean + relu ----------------
    run_sage(P.x_a, P.x_b, /*ab*/0, xb1, 1.0f, 0, 1);   // x_b1 = relu(sage_ab)
    run_sage(P.x_b, P.x_a, /*ba*/1, xa1, 0.5f, 0, 0);   // xa1  = 0.5*sage_ba
    run_sage(P.x_a, P.x_a, /*aa*/2, xa1, 0.5f, 1, 1);   // xa1  = relu(xa1 + 0.5*sage_aa)

    // ---------------- layer 1: GATv2 + hetero-mean + relu -------------------
    zero_f32_kernel<<<2 * gND, TPB, 0, stream>>>(out_a, 2 * Nn * Dd);
    run_gat(xa1, xb1, /*ab*/0, out_b, 1.0f);
    run_gat(xb1, xa1, /*ba*/1, out_a, 0.5f);
    run_gat(xa1, xa1, /*aa*/2, out_a, 0.5f);
    finalize_kernel<<<gND, TPB, 0, stream>>>(out_a, P.gb[1], 0.5f, P.gb[2], 0.5f, Nn * Dd);
    finalize_kernel<<<gND, TPB, 0, stream>>>(out_b, P.gb[0], 1.0f, nullptr, 0.0f, Nn * Dd);
}